// LearnableUpsampler_89060441850449
// MI455X (gfx1250) — compile-verified
//
#include <hip/hip_runtime.h>
#include <math.h>

// LearnableUpsampler, fused for MI455X (gfx1250, wave32, WMMA).
// Heavy math (17 GFLOP fp32) runs on V_WMMA_F32_16X16X4_F32.

#define B_  16
#define D_  256
#define L_  256
#define T_  1000
#define TP_ 1008          // T padded to multiple of 16 for WMMA tiles
#define NT_ (TP_ / 16)    // 63 tiles
#define K_  3
#define DW_ 4
#define DC_ 2
#define CH_ 8

typedef float v2f __attribute__((ext_vector_type(2)));
typedef float v8f __attribute__((ext_vector_type(8)));

// ---- workspace layout (float indices) ----
constexpr size_t WS_SK    = 0;                                  // B*L
constexpr size_t WS_BASEW = WS_SK    + (size_t)B_ * L_;         // B*L*DW
constexpr size_t WS_BASEC = WS_BASEW + (size_t)B_ * L_ * DW_;   // B*L*DC
constexpr size_t WS_FLEN  = WS_BASEC + (size_t)B_ * L_ * DC_;   // B ints
constexpr size_t WS_AFOLD = WS_FLEN  + 16;                      // 2*CH*D*K
constexpr size_t WS_BB    = WS_AFOLD + (size_t)2 * CH_ * D_ * K_; // 2*CH*K
constexpr size_t WS_W     = WS_BB    + (size_t)2 * CH_ * K_;    // B*DW*TP*L  (even -> 8B aligned)

__device__ __forceinline__ float swish_f(float x) { return x / (1.f + __expf(-x)); }

struct f4s { float a, b, c, d; };
__device__ __forceinline__ f4s f4max(f4s x, f4s y) {
  return { fmaxf(x.a, y.a), fmaxf(x.b, y.b), fmaxf(x.c, y.c), fmaxf(x.d, y.d) };
}
__device__ __forceinline__ f4s f4add(f4s x, f4s y) {
  return { x.a + y.a, x.b + y.b, x.c + y.c, x.d + y.d };
}

// ---------------------------------------------------------------------------
// Kernel 0: fold projection into conv weights.
//   Afold[br][ch][d'][k] = sum_d convW[ch,d,k] * projW[d,d']
//   Bb[br][ch][k]        = sum_d convW[ch,d,k] * projb[d]
// grid: 2*CH*K blocks of 256 (thread = d')
// ---------------------------------------------------------------------------
__global__ void k_fold(const float* __restrict__ projW_w, const float* __restrict__ convW_w,
                       const float* __restrict__ projb_w,
                       const float* __restrict__ projW_c, const float* __restrict__ convW_c,
                       const float* __restrict__ projb_c,
                       float* __restrict__ ws) {
  const int blk = blockIdx.x;
  const int br  = blk / (CH_ * K_);
  const int rem = blk % (CH_ * K_);
  const int ch  = rem / K_;
  const int k   = rem % K_;
  const float* projW = br ? projW_c : projW_w;
  const float* convW = br ? convW_c : convW_w;
  const float* projb = br ? projb_c : projb_w;
  const int dp = threadIdx.x;

  float acc = 0.f;
  for (int d = 0; d < D_; ++d)
    acc = fmaf(convW[(ch * D_ + d) * K_ + k], projW[d * D_ + dp], acc);
  ws[WS_AFOLD + ((size_t)(br * CH_ + ch) * D_ + dp) * K_ + k] = acc;

  __shared__ float red[256];
  red[dp] = convW[(ch * D_ + dp) * K_ + k] * projb[dp];
  __syncthreads();
  for (int s = 128; s > 0; s >>= 1) {
    if (dp < s) red[dp] += red[dp + s];
    __syncthreads();
  }
  if (dp == 0) ws[WS_BB + (size_t)(br * CH_ + ch) * K_ + k] = red[0];
}

// ---------------------------------------------------------------------------
// Kernel 1: per batch: cumsum(durations) -> sk, frame_len; folded conv ->
// h_w/h_c; t-independent MLP bases base_w[l][q], base_c[l][p].
// grid: B blocks of 256 (thread = l)
// ---------------------------------------------------------------------------
__global__ void k_setup(const float* __restrict__ dur_g, const float* __restrict__ ph,
                        const float* __restrict__ mwW, const float* __restrict__ mwb,
                        const float* __restrict__ cwb_w,
                        const float* __restrict__ mcW, const float* __restrict__ mcb,
                        const float* __restrict__ cwb_c,
                        float* __restrict__ ws) {
  const int b = blockIdx.x;
  const int l = threadIdx.x;

  __shared__ float sc[256];
  __shared__ float Alds[2 * CH_ * D_ * K_];   // 48 KB
  for (int i = l; i < 2 * CH_ * D_ * K_; i += 256) Alds[i] = ws[WS_AFOLD + i];

  const float dur = dur_g[b * L_ + l];
  const bool  pad = (dur == 0.f);             // phoneme_mask positions have dur==0

  // inclusive scan (Hillis-Steele)
  sc[l] = dur;
  __syncthreads();
  for (int off = 1; off < 256; off <<= 1) {
    float v = (l >= off) ? sc[l - off] : 0.f;
    __syncthreads();
    sc[l] += v;
    __syncthreads();
  }
  const float sd = sc[l];
  const float sk = sd - dur;
  ws[WS_SK + b * L_ + l] = sk;
  if (l == L_ - 1) {
    int fl = (int)rintf(sd);
    fl = fl < 0 ? 0 : (fl > T_ ? T_ : fl);
    ((int*)ws)[WS_FLEN + b] = fl;
  }

  // folded conv: h[ch,l] = sum_{d,k} phoneme[d, l+k-1] * A[ch,d,k]
  float hw[CH_], hc[CH_];
#pragma unroll
  for (int ch = 0; ch < CH_; ++ch) { hw[ch] = 0.f; hc[ch] = 0.f; }
  const float* phb = ph + (size_t)b * D_ * L_;
  for (int d = 0; d < D_; ++d) {
    const float pm = (l > 0)      ? phb[d * L_ + l - 1] : 0.f;
    const float p0 =                phb[d * L_ + l];
    const float pp = (l < L_ - 1) ? phb[d * L_ + l + 1] : 0.f;
#pragma unroll
    for (int ch = 0; ch < CH_; ++ch) {
      const float* Aw = &Alds[((0 * CH_ + ch) * D_ + d) * K_];
      const float* Ac = &Alds[((1 * CH_ + ch) * D_ + d) * K_];
      hw[ch] = fmaf(pm, Aw[0], fmaf(p0, Aw[1], fmaf(pp, Aw[2], hw[ch])));
      hc[ch] = fmaf(pm, Ac[0], fmaf(p0, Ac[1], fmaf(pp, Ac[2], hc[ch])));
    }
  }
#pragma unroll
  for (int ch = 0; ch < CH_; ++ch) {
    float bbw = 0.f, bbc = 0.f;
#pragma unroll
    for (int k = 0; k < K_; ++k) {
      const int ll = l + k - 1;
      if (ll >= 0 && ll < L_) {
        bbw += ws[WS_BB + (0 * CH_ + ch) * K_ + k];
        bbc += ws[WS_BB + (1 * CH_ + ch) * K_ + k];
      }
    }
    hw[ch] = pad ? 0.f : swish_f(hw[ch] + bbw + cwb_w[ch]);
    hc[ch] = pad ? 0.f : swish_f(hc[ch] + bbc + cwb_c[ch]);
  }

  // base_w[l][q] = E*mwW[1][q] + sum_ch h_w[ch]*mwW[2+ch][q] + mwb[q]   (E = dur)
#pragma unroll
  for (int q = 0; q < DW_; ++q) {
    float v = fmaf(dur, mwW[1 * DW_ + q], mwb[q]);
#pragma unroll
    for (int ch = 0; ch < CH_; ++ch) v = fmaf(hw[ch], mwW[(2 + ch) * DW_ + q], v);
    ws[WS_BASEW + ((size_t)b * L_ + l) * DW_ + q] = v;
  }
#pragma unroll
  for (int p = 0; p < DC_; ++p) {
    float v = fmaf(dur, mcW[1 * DC_ + p], mcb[p]);
#pragma unroll
    for (int ch = 0; ch < CH_; ++ch) v = fmaf(hc[ch], mcW[(2 + ch) * DC_ + p], v);
    ws[WS_BASEC + ((size_t)b * L_ + l) * DC_ + p] = v;
  }
}

// ---------------------------------------------------------------------------
// Kernel 2: per (b,t): attention weights w (softmax over L), write w to ws;
// fold the tiny wc path + biases directly into out.
// grid: B*TP blocks of 256 (thread = l, doubles as d2 for the output row)
// ---------------------------------------------------------------------------
__global__ void k_attn(const float* __restrict__ dur_g,
                       const float* __restrict__ mwW, const float* __restrict__ mcW,
                       const float* __restrict__ linc_W,
                       const float* __restrict__ linw_b, const float* __restrict__ linc_b,
                       float* __restrict__ ws, float* __restrict__ out) {
  const int bx = blockIdx.x;
  const int b  = bx / TP_;
  const int t  = bx % TP_;
  const int l  = threadIdx.x;
  const int fl = ((const int*)ws)[WS_FLEN + b];

  if (t >= fl) {  // frame padding (and TP padding): w row = 0, out = biases
#pragma unroll
    for (int q = 0; q < DW_; ++q)
      ws[WS_W + ((size_t)(b * DW_ + q) * TP_ + t) * L_ + l] = 0.f;
    if (t < T_) out[((size_t)b * T_ + t) * D_ + l] = linw_b[l] + linc_b[l];
    return;
  }

  const float dur = dur_g[b * L_ + l];
  const bool  pad = (dur == 0.f);
  const float sk  = ws[WS_SK + b * L_ + l];
  const float S   = (float)(t + 1) - sk;

  float a[DW_];
#pragma unroll
  for (int q = 0; q < DW_; ++q)
    a[q] = pad ? -__builtin_inff()
               : swish_f(fmaf(S, mwW[q], ws[WS_BASEW + ((size_t)b * L_ + l) * DW_ + q]));

  __shared__ f4s red[256];
  // max over l (4 lanes at once)
  red[l] = { a[0], a[1], a[2], a[3] };
  __syncthreads();
  for (int s = 128; s > 0; s >>= 1) {
    if (l < s) red[l] = f4max(red[l], red[l + s]);
    __syncthreads();
  }
  const f4s m = red[0];
  __syncthreads();
  float e[DW_];
  e[0] = pad ? 0.f : __expf(a[0] - m.a);
  e[1] = pad ? 0.f : __expf(a[1] - m.b);
  e[2] = pad ? 0.f : __expf(a[2] - m.c);
  e[3] = pad ? 0.f : __expf(a[3] - m.d);
  red[l] = { e[0], e[1], e[2], e[3] };
  __syncthreads();
  for (int s = 128; s > 0; s >>= 1) {
    if (l < s) red[l] = f4add(red[l], red[l + s]);
    __syncthreads();
  }
  const f4s sum = red[0];
  __syncthreads();
  float wv[DW_];
  wv[0] = e[0] / sum.a; wv[1] = e[1] / sum.b; wv[2] = e[2] / sum.c; wv[3] = e[3] / sum.d;
#pragma unroll
  for (int q = 0; q < DW_; ++q)
    ws[WS_W + ((size_t)(b * DW_ + q) * TP_ + t) * L_ + l] = wv[q];

  // c[l][p] = swish(S*mcW[p] + base_c); value at pad lanes irrelevant (wv=0)
  float c0 = swish_f(fmaf(S, mcW[0], ws[WS_BASEC + ((size_t)b * L_ + l) * DC_ + 0]));
  float c1 = swish_f(fmaf(S, mcW[1], ws[WS_BASEC + ((size_t)b * L_ + l) * DC_ + 1]));

  // wc[q][p] = sum_l wv[q]*c[p]  (8 values, two f4 reductions)
  red[l] = { wv[0] * c0, wv[0] * c1, wv[1] * c0, wv[1] * c1 };
  __syncthreads();
  for (int s = 128; s > 0; s >>= 1) {
    if (l < s) red[l] = f4add(red[l], red[l + s]);
    __syncthreads();
  }
  const f4s wc01 = red[0];
  __syncthreads();
  red[l] = { wv[2] * c0, wv[2] * c1, wv[3] * c0, wv[3] * c1 };
  __syncthreads();
  for (int s = 128; s > 0; s >>= 1) {
    if (l < s) red[l] = f4add(red[l], red[l + s]);
    __syncthreads();
  }
  const f4s wc23 = red[0];

  if (t < T_) {
    float o = linw_b[l] + linc_b[l];
    o = fmaf(wc01.a, linc_W[0 * D_ + l], o);
    o = fmaf(wc01.b, linc_W[1 * D_ + l], o);
    o = fmaf(wc01.c, linc_W[2 * D_ + l], o);
    o = fmaf(wc01.d, linc_W[3 * D_ + l], o);
    o = fmaf(wc23.a, linc_W[4 * D_ + l], o);
    o = fmaf(wc23.b, linc_W[5 * D_ + l], o);
    o = fmaf(wc23.c, linc_W[6 * D_ + l], o);
    o = fmaf(wc23.d, linc_W[7 * D_ + l], o);
    out[((size_t)b * T_ + t) * D_ + l] = o;
  }
}

// ---------------------------------------------------------------------------
// Kernel 3: WMMA double GEMM, fused per (b, 16-t tile).
//   For q = 0..3:  WH_q(16x256) = w_q(16xL) * P^T(LxD)   -> LDS
//                  oacc(16x256) += WH_q * lin_w_W[q*256:(q+1)*256, :]
//   out += oacc
// grid: B*NT blocks of 128 (4 waves; each wave owns 64 output columns)
// ---------------------------------------------------------------------------
__global__ void __launch_bounds__(128)
k_wmma(const float* __restrict__ ph, const float* __restrict__ linw_W,
       const float* __restrict__ ws, float* __restrict__ out) {
  constexpr int PITCH = 260;                  // bank stride 4 -> conflict free rows
  __shared__ float WHt[16 * PITCH];           // ~16.6 KB

  const int bx   = blockIdx.x;
  const int b    = bx / NT_;
  const int t0   = (bx % NT_) * 16;
  const int tid  = threadIdx.x;
  const int wave = tid >> 5;
  const int lane = tid & 31;
  const int lo   = lane & 15;                 // M row / N col within 16-tile
  const int hi   = lane >> 4;                 // K sub-pair selector

  const float* phb = ph + (size_t)b * D_ * L_;

  v8f oacc[4] = { {}, {}, {}, {} };           // 16t x 64d2 per wave

  for (int q = 0; q < DW_; ++q) {
    // ---- GEMM1: WH_q tile ----
    v8f acc[4] = { {}, {}, {}, {} };
    const float* wrow = ws + WS_W + ((size_t)(b * DW_ + q) * TP_ + (t0 + lo)) * L_;
    for (int k0 = 0; k0 < L_; k0 += 4) {
      const v2f a = *(const v2f*)(wrow + k0 + hi * 2);          // A[lo][k0+2hi .. +1]
#pragma unroll
      for (int j = 0; j < 4; ++j) {
        const int n = wave * 64 + j * 16 + lo;
        const v2f bb = *(const v2f*)(phb + (size_t)n * L_ + k0 + hi * 2); // B[k][n]=P[n][k]
        acc[j] = __builtin_amdgcn_wmma_f32_16x16x4_f32(
            false, a, false, bb, (short)0, acc[j], false, false);
      }
    }
#pragma unroll
    for (int j = 0; j < 4; ++j)
#pragma unroll
      for (int r = 0; r < 8; ++r)
        WHt[(r + hi * 8) * PITCH + wave * 64 + j * 16 + lo] = acc[j][r];
    __syncthreads();

    // ---- GEMM2 partial: oacc += WH_q * lin_w_W rows [q*256, q*256+256) ----
    const float* lw = linw_W + (size_t)q * 256 * D_;
    for (int k0 = 0; k0 < 256; k0 += 4) {
      const int kk = k0 + hi * 2;
      const v2f a = *(const v2f*)(&WHt[lo * PITCH + kk]);
#pragma unroll
      for (int j = 0; j < 4; ++j) {
        const int n = wave * 64 + j * 16 + lo;
        v2f bb;
        bb.x = lw[(size_t)kk * D_ + n];
        bb.y = lw[(size_t)(kk + 1) * D_ + n];
        oacc[j] = __builtin_amdgcn_wmma_f32_16x16x4_f32(
            false, a, false, bb, (short)0, oacc[j], false, false);
      }
    }
    __syncthreads();  // protect WHt before next q overwrites it
  }

  // out += oacc  (C/D layout: VGPR r -> M = r + 8*hi, N = lo)
#pragma unroll
  for (int j = 0; j < 4; ++j)
#pragma unroll
    for (int r = 0; r < 8; ++r) {
      const int trow = t0 + r + hi * 8;
      if (trow < T_) {
        const int n = wave * 64 + j * 16 + lo;
        float* op = out + ((size_t)b * T_ + trow) * D_ + n;
        *op = *op + oacc[j][r];
      }
    }
}

// ---------------------------------------------------------------------------
extern "C" void kernel_launch(void* const* d_in, const int* in_sizes, int n_in,
                              void* d_out, int out_size, void* d_ws, size_t ws_size,
                              hipStream_t stream) {
  const float* durations = (const float*)d_in[0];
  const float* phoneme   = (const float*)d_in[1];
  // d_in[2] phoneme_mask: recovered exactly as (durations == 0), avoids bool dtype
  const float* proj_w_W  = (const float*)d_in[3];
  const float* proj_w_b  = (const float*)d_in[4];
  const float* conv_w_W  = (const float*)d_in[5];
  const float* conv_w_b  = (const float*)d_in[6];
  const float* mlp_w_W   = (const float*)d_in[7];
  const float* mlp_w_b   = (const float*)d_in[8];
  const float* lin_w_W   = (const float*)d_in[9];
  const float* lin_w_b   = (const float*)d_in[10];
  const float* proj_c_W  = (const float*)d_in[11];
  const float* proj_c_b  = (const float*)d_in[12];
  const float* conv_c_W  = (const float*)d_in[13];
  const float* conv_c_b  = (const float*)d_in[14];
  const float* mlp_c_W   = (const float*)d_in[15];
  const float* mlp_c_b   = (const float*)d_in[16];
  const float* lin_c_W   = (const float*)d_in[17];
  const float* lin_c_b   = (const float*)d_in[18];

  float* out = (float*)d_out;
  float* ws  = (float*)d_ws;

  k_fold<<<2 * CH_ * K_, 256, 0, stream>>>(proj_w_W, conv_w_W, proj_w_b,
                                           proj_c_W, conv_c_W, proj_c_b, ws);
  k_setup<<<B_, 256, 0, stream>>>(durations, phoneme, mlp_w_W, mlp_w_b, conv_w_b,
                                  mlp_c_W, mlp_c_b, conv_c_b, ws);
  k_attn<<<B_ * TP_, 256, 0, stream>>>(durations, mlp_w_W, mlp_c_W, lin_c_W,
                                       lin_w_b, lin_c_b, ws, out);
  k_wmma<<<B_ * NT_, 128, 0, stream>>>(phoneme, lin_w_W, ws, out);
}